// HypergraphConv_10187662426196
// MI455X (gfx1250) — compile-verified
//
#include <hip/hip_runtime.h>
#include <hip/hip_bf16.h>

// ---------------------------------------------------------------------------
// HypergraphConv for MI455X (gfx1250, wave32)
//   H = X@W + b          (fp32 WMMA 16x16x4)
//   Y = segmean_e(H[vidx])   via atomic float scatter-add + normalize
//   O = relu(segmean_v(Y[eidx]))
// ---------------------------------------------------------------------------

typedef __attribute__((ext_vector_type(2))) float v2f;
typedef __attribute__((ext_vector_type(8))) float v8f;

#define D 256                    // D_IN == D_OUT == 256
static constexpr int N_NODES = 100000;
static constexpr int N_EDGES = 20000;

// ---------------------------------------------------------------------------
// Kernel 1: H = X @ W + b using V_WMMA_F32_16X16X4_F32.
// One wave (32 lanes) computes one 16x16 output tile; K marched in steps of 4.
// A layout (16x4 fp32, ISA 7.12.2): lanes 0-15 hold K=0 (v0), K=1 (v1);
//                                   lanes 16-31 hold K=2 (v0), K=3 (v1).
// B layout (4x16 fp32): VGPR v, lane L -> K = v + 2*(L>>4), N = L&15.
// C/D layout (16x16 fp32): VGPR v, lane L -> M = v + 8*(L>>4), N = L&15.
// ---------------------------------------------------------------------------
__global__ void hg_gemm_bias_wmma(const float* __restrict__ X,
                                  const float* __restrict__ W,
                                  const float* __restrict__ bias,
                                  float* __restrict__ H,
                                  int M) {
  const int wavesPerBlock = blockDim.x >> 5;
  const int wave = blockIdx.x * wavesPerBlock + (threadIdx.x >> 5);
  const int lane = threadIdx.x & 31;
  const int tilesN = D / 16;               // 16 column tiles
  const int tm = wave / tilesN;
  const int tn = wave - tm * tilesN;
  if (tm * 16 >= M) return;

  const int m0 = tm * 16;
  const int n0 = tn * 16;
  const int lo = lane & 15;                // row (A/C) or col (B/C)
  const int hi = lane >> 4;                // selects K/M half

  const float* __restrict__ arow = X + (size_t)(m0 + lo) * D;
  const float* __restrict__ bcol = W + (size_t)n0 + lo;

  v8f c = {};
  #pragma unroll 8
  for (int k0 = 0; k0 < D; k0 += 4) {
    const int ka = k0 + 2 * hi;
    v2f a, b;
    a.x = arow[ka + 0];
    a.y = arow[ka + 1];
    b.x = bcol[(size_t)(ka + 0) * D];
    b.y = bcol[(size_t)(ka + 1) * D];
    // 8 args: (neg_a, A, neg_b, B, c_mod, C, reuse_a, reuse_b)
    c = __builtin_amdgcn_wmma_f32_16x16x4_f32(false, a, false, b,
                                              (short)0, c, false, false);
  }

  const float bb = bias[n0 + lo];
  float* __restrict__ hout = H + (size_t)n0 + lo;
  #pragma unroll
  for (int v = 0; v < 8; ++v) {
    const int row = m0 + v + 8 * hi;
    hout[(size_t)row * D] = c[v] + bb;
  }
}

// ---------------------------------------------------------------------------
// Zero a float region (grid-stride). Needed because d_out/d_ws are poisoned.
// ---------------------------------------------------------------------------
__global__ void hg_zero(float* __restrict__ p, size_t n) {
  size_t i = (size_t)blockIdx.x * blockDim.x + threadIdx.x;
  size_t stride = (size_t)gridDim.x * blockDim.x;
  for (; i < n; i += stride) p[i] = 0.0f;
}

// ---------------------------------------------------------------------------
// Scatter-add one gathered row per nnz entry: 64 threads/entry, float4 each.
// Also bumps the per-segment count (thread part==0).
// ---------------------------------------------------------------------------
__global__ void hg_scatter(const float* __restrict__ src,   // [nsrc, 256]
                           const int* __restrict__ gather_idx,
                           const int* __restrict__ segment_idx,
                           float* __restrict__ seg_sum,     // [nseg, 256]
                           float* __restrict__ seg_cnt,     // [nseg]
                           int nnz) {
  const long long t = (long long)blockIdx.x * blockDim.x + threadIdx.x;
  const int entry = (int)(t >> 6);
  const int part  = (int)(t & 63);
  if (entry >= nnz) return;

  const int g = gather_idx[entry];
  const int s = segment_idx[entry];

  const float4 val = ((const float4*)(src + (size_t)g * D))[part];
  float* dst = seg_sum + (size_t)s * D + (size_t)part * 4;
  atomicAdd(dst + 0, val.x);
  atomicAdd(dst + 1, val.y);
  atomicAdd(dst + 2, val.z);
  atomicAdd(dst + 3, val.w);
  if (part == 0) atomicAdd(seg_cnt + s, 1.0f);
}

// ---------------------------------------------------------------------------
// In-place divide rows by max(count,1); optional ReLU for the final stage.
// One block (256 threads) per row.
// ---------------------------------------------------------------------------
template <bool RELU>
__global__ void hg_normalize(float* __restrict__ rows,
                             const float* __restrict__ cnt,
                             int nrows) {
  const int r = blockIdx.x;
  if (r >= nrows) return;
  const float inv = 1.0f / fmaxf(cnt[r], 1.0f);
  float v = rows[(size_t)r * D + threadIdx.x] * inv;
  if (RELU) v = fmaxf(v, 0.0f);
  rows[(size_t)r * D + threadIdx.x] = v;
}

// ---------------------------------------------------------------------------
// Host-side launcher (graph-capture safe: only kernel launches on `stream`).
// Inputs: [0]=X [100000,256] f32, [1]=W [256,256] f32, [2]=b [256] f32,
//         [3]=vertex_idx [800000] i32, [4]=edge_idx [800000] i32.
// Output: [100000,256] f32.
// Workspace layout (bytes):
//   H:     [0, 100000*256*4)                          = 102,400,000
//   Ysum:  next 20000*256*4                           =  20,480,000
//   ecnt:  next 20000*4
//   vcnt:  next 100000*4
// ---------------------------------------------------------------------------
extern "C" void kernel_launch(void* const* d_in, const int* in_sizes, int n_in,
                              void* d_out, int out_size, void* d_ws, size_t ws_size,
                              hipStream_t stream) {
  const float* X    = (const float*)d_in[0];
  const float* W    = (const float*)d_in[1];
  const float* bias = (const float*)d_in[2];
  const int* vidx   = (const int*)d_in[3];
  const int* eidx   = (const int*)d_in[4];
  float* out        = (float*)d_out;

  const int M   = in_sizes[0] / D;   // 100000
  const int nnz = in_sizes[3];       // 800000

  char* ws = (char*)d_ws;
  float* H    = (float*)ws;
  float* Ysum = (float*)(ws + (size_t)M * D * sizeof(float));
  float* ecnt = (float*)((char*)Ysum + (size_t)N_EDGES * D * sizeof(float));
  float* vcnt = (float*)((char*)ecnt + (size_t)N_EDGES * sizeof(float));

  // 1) Zero accumulators: Ysum+ecnt+vcnt are contiguous; out separately.
  const size_t nzero_ws = (size_t)N_EDGES * D + N_EDGES + M;
  hg_zero<<<2048, 256, 0, stream>>>(Ysum, nzero_ws);
  hg_zero<<<2048, 256, 0, stream>>>(out, (size_t)M * D);

  // 2) H = X@W + b   (WMMA fp32). 16x16 tiles: (M/16)*16 waves, 8 waves/block.
  {
    const int waves  = (M / 16) * (D / 16);      // 100000
    const int blocks = (waves + 7) / 8;          // 12500
    hg_gemm_bias_wmma<<<blocks, 256, 0, stream>>>(X, W, bias, H, M);
  }

  // 3) v2e: Ysum[e] += H[v], ecnt[e] += 1
  {
    const long long threads = (long long)nnz * 64;
    const int blocks = (int)((threads + 255) / 256); // 200000
    hg_scatter<<<blocks, 256, 0, stream>>>(H, vidx, eidx, Ysum, ecnt, nnz);
  }

  // 4) Y = Ysum / max(ecnt,1)   (in place)
  hg_normalize<false><<<N_EDGES, 256, 0, stream>>>(Ysum, ecnt, N_EDGES);

  // 5) e2v: out[v] += Y[e], vcnt[v] += 1
  {
    const long long threads = (long long)nnz * 64;
    const int blocks = (int)((threads + 255) / 256);
    hg_scatter<<<blocks, 256, 0, stream>>>(Ysum, eidx, vidx, out, vcnt, nnz);
  }

  // 6) out = relu(out / max(vcnt,1))
  hg_normalize<true><<<M, 256, 0, stream>>>(out, vcnt, M);
}